// ResearchValidatedDiffusionModel_50809463111709
// MI455X (gfx1250) — compile-verified
//
#include <hip/hip_runtime.h>
#include <cstdint>
#include <cstddef>

// ---------------- problem constants (from reference) ----------------
#define HDIM   64
#define NNODES 50000
#define NEDGES 800000

// ---------------- CDNA5 wave32 / WMMA types ----------------
typedef __attribute__((ext_vector_type(2))) float v2f;   // A/B frag of V_WMMA_F32_16X16X4_F32
typedef __attribute__((ext_vector_type(8))) float v8f;   // 16x16 f32 C/D tile

constexpr int AS    = 132;  // LDS stride of 16x(up to 128) A staging panel (16B aligned rows)
constexpr int TS    = 68;   // LDS stride of 16x64 intermediate panels
constexpr int WAVES = 4;    // 128-thread block = 4 wave32

// Intra-wave LDS producer->consumer ordering (wave32 lockstep; DS ops are
// in-order per wave). The "memory" clobber also blocks compiler reordering.
__device__ __forceinline__ void lds_fence() {
  asm volatile("s_wait_dscnt 0" ::: "memory");
}

__device__ __forceinline__ float silu_f(float x) {
  return x * (1.0f / (1.0f + __expf(-x)));
}

__device__ __forceinline__ float tanh_f(float x) {
  float t = __expf(-2.0f * fabsf(x));
  float r = (1.0f - t) / (1.0f + t);
  return copysignf(r, x);
}

// Weight repack: row-major W[K][64]  ->  k-pair interleaved Wp[(k/2)*128 + n*2 + (k&1)].
// B fragment for (k, half, ncol) is then the contiguous 8B pair at
// Wp + (k/2 + half)*128 + ncol*2  -> single ds_load_b64 into an aligned VGPR pair.
__device__ __forceinline__ void repack_pairs(float* __restrict__ dst,
                                             const float* __restrict__ src,
                                             int kRows, int tid, int nthreads) {
  for (int i = tid; i < kRows * HDIM; i += nthreads) {
    const int k = i >> 6;
    const int n = i & 63;
    dst[(k >> 1) * 128 + n * 2 + (k & 1)] = src[i];
  }
}

// One wave computes D[16 x 64] = A[16 x K] * Wp[K x 64] + bias using
// V_WMMA_F32_16X16X4_F32. A is an LDS panel (row m at A[m*As]); Wp is the
// pair-interleaved weight block. acc[nt] holds output columns nt*16..nt*16+15.
// Fragment mapping (ISA 7.12.2, 32-bit 16x4 A): lanes 0-15 carry K=(k,k+1),
// lanes 16-31 carry K=(k+2,k+3); C/D: vgpr r = row (r + 8*half), col = lane&15.
template <int K>
__device__ __forceinline__ void wave_gemm_16xK_x64(const float* __restrict__ A, int As,
                                                   const float* __restrict__ Wp,
                                                   const float* __restrict__ bias,
                                                   v8f acc[4], int lrow, int half) {
#pragma unroll
  for (int nt = 0; nt < 4; ++nt) {
    float b = bias[nt * 16 + lrow];
    v8f t = {b, b, b, b, b, b, b, b};
    acc[nt] = t;
  }
#pragma unroll
  for (int k = 0; k < K; k += 4) {
    const v2f a = *reinterpret_cast<const v2f*>(A + lrow * As + k + 2 * half);
    const float* wrow = Wp + ((k >> 1) + half) * 128;
#pragma unroll
    for (int nt = 0; nt < 4; ++nt) {
      const v2f b = *reinterpret_cast<const v2f*>(wrow + (nt * 16 + lrow) * 2);
      acc[nt] = __builtin_amdgcn_wmma_f32_16x16x4_f32(
          /*neg_a=*/false, a, /*neg_b=*/false, b,
          /*c_mod=*/(short)0, acc[nt], /*reuse_a=*/false, /*reuse_b=*/false);
    }
  }
}

// Store a 16x64 C/D tile from acc[] into an LDS panel, optionally applying SiLU.
__device__ __forceinline__ void store_tile(float* __restrict__ T, int Ts, const v8f acc[4],
                                           int lrow, int half, bool doSilu) {
#pragma unroll
  for (int nt = 0; nt < 4; ++nt) {
#pragma unroll
    for (int r = 0; r < 8; ++r) {
      float v = acc[nt][r];
      if (doSilu) v = silu_f(v);
      T[(r + 8 * half) * Ts + nt * 16 + lrow] = v;
    }
  }
}

// LayerNorm one 64-wide row in LDS, in place. Called by lanes 0..15 (one row each).
__device__ __forceinline__ void ln_row_lds(float* __restrict__ row,
                                           const float* __restrict__ g,
                                           const float* __restrict__ be) {
  float m = 0.f;
#pragma unroll
  for (int n = 0; n < HDIM; ++n) m += row[n];
  m *= (1.0f / HDIM);
  float v = 0.f;
#pragma unroll
  for (int n = 0; n < HDIM; ++n) {
    float d = row[n] - m;
    v += d * d;
  }
  v *= (1.0f / HDIM);
  float rs = rsqrtf(v + 1e-5f);
#pragma unroll
  for (int n = 0; n < HDIM; ++n) row[n] = (row[n] - m) * rs * g[n] + be[n];
}

// ------------------------------------------------------------------
// Kernel 0: zero an fp32 buffer (accumulators must start at 0).
// ------------------------------------------------------------------
__global__ void egnn_zero_kernel(float* __restrict__ p, size_t n) {
  size_t i = (size_t)blockIdx.x * blockDim.x + threadIdx.x;
  if (i < n) p[i] = 0.0f;
}

// ------------------------------------------------------------------
// Kernel 1: per-edge MLPs + scatter. One wave per 16-edge tile.
// ------------------------------------------------------------------
__global__ __launch_bounds__(128) void egnn_edge_kernel(
    const float* __restrict__ h, const float* __restrict__ pos,
    const int* __restrict__ eidx,
    const float* __restrict__ w1, const float* __restrict__ b1,
    const float* __restrict__ w2, const float* __restrict__ b2,
    const float* __restrict__ g, const float* __restrict__ be,
    const float* __restrict__ wx1, const float* __restrict__ bx1,
    const float* __restrict__ wx2, const float* __restrict__ bx2,
    float* __restrict__ hmsg, float* __restrict__ posOut) {
  // Weights resident in LDS for the whole block (~67 KB of the 320 KB WGP pool),
  // pair-interleaved for single-b64 B-fragment loads.
  __shared__ __align__(16) float sW1p[128 * HDIM];   // rows 0..127, paired
  __shared__ __align__(16) float sW1last[HDIM];      // row 128 (dist column)
  __shared__ __align__(16) float sW2p[HDIM * HDIM];
  __shared__ __align__(16) float sWx1p[HDIM * HDIM];
  __shared__ float sB1[HDIM];
  __shared__ float sB2[HDIM];
  __shared__ float sG[HDIM];
  __shared__ float sBe[HDIM];
  __shared__ float sBx1[HDIM];
  __shared__ float sWx2[HDIM];
  __shared__ float sBx2;
  // Per-wave staging panels.
  __shared__ __align__(16) float sAbuf[WAVES][16 * AS];
  __shared__ __align__(16) float sTbuf[WAVES][16 * TS];
  __shared__ float sRp[WAVES][16][4];
  __shared__ float sDist[WAVES][16];
  __shared__ int sCol[WAVES][16];

  const int tid = threadIdx.x;
  repack_pairs(sW1p, w1, 128, tid, blockDim.x);
  repack_pairs(sW2p, w2, HDIM, tid, blockDim.x);
  repack_pairs(sWx1p, wx1, HDIM, tid, blockDim.x);
  for (int i = tid; i < HDIM; i += blockDim.x) {
    sW1last[i] = w1[128 * HDIM + i];
    sB1[i] = b1[i]; sB2[i] = b2[i]; sG[i] = g[i]; sBe[i] = be[i];
    sBx1[i] = bx1[i]; sWx2[i] = wx2[i];
  }
  if (tid == 0) sBx2 = bx2[0];
  __syncthreads();

  const int lane = tid & 31;
  const int w    = tid >> 5;
  const int lrow = lane & 15;
  const int half = lane >> 4;
  float* Abuf = sAbuf[w];
  float* Tbuf = sTbuf[w];

  const int nTiles = NEDGES / 16;           // 50000, exact
  const int slots  = gridDim.x * WAVES;     // wave-uniform

  for (int tile = blockIdx.x * WAVES + w; tile < nTiles; tile += slots) {
    // ---- gather: lanes (lrow,half) stage h[row] (half=0) / h[col] (half=1) ----
    const int e = tile * 16 + lrow;
    int rr = eidx[e];          rr = rr < 0 ? 0 : (rr >= NNODES ? NNODES - 1 : rr);
    int cc = eidx[NEDGES + e]; cc = cc < 0 ? 0 : (cc >= NNODES ? NNODES - 1 : cc);
    const int src = half ? cc : rr;
    const float4* hp = reinterpret_cast<const float4*>(h + (size_t)src * HDIM);
#pragma unroll
    for (int k4 = 0; k4 < 16; ++k4) {
      float4 v = hp[k4];
      float* d = Abuf + lrow * AS + half * 64 + k4 * 4;   // 16B-aligned -> ds_store_b128
      d[0] = v.x; d[1] = v.y; d[2] = v.z; d[3] = v.w;
    }
    if (!half) {
      const float px = pos[(size_t)rr * 3 + 0] - pos[(size_t)cc * 3 + 0];
      const float py = pos[(size_t)rr * 3 + 1] - pos[(size_t)cc * 3 + 1];
      const float pz = pos[(size_t)rr * 3 + 2] - pos[(size_t)cc * 3 + 2];
      sRp[w][lrow][0] = px; sRp[w][lrow][1] = py; sRp[w][lrow][2] = pz;
      const float ax = px + 1e-8f, ay = py + 1e-8f, az = pz + 1e-8f; // reference adds 1e-8 pre-norm
      sDist[w][lrow] = sqrtf(ax * ax + ay * ay + az * az);
      sCol[w][lrow] = cc;
    }
    lds_fence();

    // ---- phi_e layer 1: [16x128]@[128x64] on WMMA + rank-1 dist column ----
    v8f acc[4];
    wave_gemm_16xK_x64<128>(Abuf, AS, sW1p, sB1, acc, lrow, half);
    float dv[8];
#pragma unroll
    for (int r = 0; r < 8; ++r) dv[r] = sDist[w][r + 8 * half];
#pragma unroll
    for (int nt = 0; nt < 4; ++nt) {
      const float wv = sW1last[nt * 16 + lrow];
#pragma unroll
      for (int r = 0; r < 8; ++r) acc[nt][r] += dv[r] * wv;
    }
    store_tile(Tbuf, TS, acc, lrow, half, /*silu=*/true);
    lds_fence();

    // ---- phi_e layer 2 + LayerNorm -> edge_msg in Tbuf ----
    wave_gemm_16xK_x64<HDIM>(Tbuf, TS, sW2p, sB2, acc, lrow, half);
    lds_fence();                     // all Tbuf reads done before overwrite
    store_tile(Tbuf, TS, acc, lrow, half, /*silu=*/false);
    lds_fence();
    if (!half) ln_row_lds(Tbuf + lrow * TS, sG, sBe);
    lds_fence();

    // ---- scatter edge_msg into h_msg[col] (coalesced f32 atomics) ----
#pragma unroll
    for (int i = 0; i < 32; ++i) {
      const int v  = i * 32 + lane;     // 0..1023 over (edge, feature)
      const int eL = v >> 6;
      const int f  = v & 63;
      atomicAdd(hmsg + (size_t)sCol[w][eL] * HDIM + f, Tbuf[eL * TS + f]);
    }

    // ---- phi_x hidden: edge_msg @ Wx1, SiLU -> Abuf ----
    wave_gemm_16xK_x64<HDIM>(Tbuf, TS, sWx1p, sBx1, acc, lrow, half);
    store_tile(Abuf, AS, acc, lrow, half, /*silu=*/true);
    lds_fence();

    // ---- phi_x output: dot with wx2, tanh, scatter rel_pos * coef ----
    if (!half) {
      float s = sBx2;
#pragma unroll
      for (int n = 0; n < HDIM; ++n) s += Abuf[lrow * AS + n] * sWx2[n];
      const float coef = tanh_f(s);
      const int cn = sCol[w][lrow];
      atomicAdd(posOut + (size_t)cn * 3 + 0, sRp[w][lrow][0] * coef);
      atomicAdd(posOut + (size_t)cn * 3 + 1, sRp[w][lrow][1] * coef);
      atomicAdd(posOut + (size_t)cn * 3 + 2, sRp[w][lrow][2] * coef);
    }
    lds_fence();  // protect panels before next iteration's staging
  }
}

// ------------------------------------------------------------------
// Kernel 2: per-node MLP. One wave per 16-node tile.
// ------------------------------------------------------------------
__global__ __launch_bounds__(128) void egnn_node_kernel(
    const float* __restrict__ h, const float* __restrict__ hmsg,
    const float* __restrict__ w1, const float* __restrict__ b1,
    const float* __restrict__ w2, const float* __restrict__ b2,
    const float* __restrict__ g, const float* __restrict__ be,
    float* __restrict__ out) {
  __shared__ __align__(16) float sW1p[HDIM * HDIM];
  __shared__ __align__(16) float sW2p[HDIM * HDIM];
  __shared__ float sB1[HDIM];
  __shared__ float sB2[HDIM];
  __shared__ float sG[HDIM];
  __shared__ float sBe[HDIM];
  __shared__ __align__(16) float sAbuf[WAVES][16 * TS];
  __shared__ __align__(16) float sTbuf[WAVES][16 * TS];

  const int tid = threadIdx.x;
  repack_pairs(sW1p, w1, HDIM, tid, blockDim.x);
  repack_pairs(sW2p, w2, HDIM, tid, blockDim.x);
  for (int i = tid; i < HDIM; i += blockDim.x) {
    sB1[i] = b1[i]; sB2[i] = b2[i]; sG[i] = g[i]; sBe[i] = be[i];
  }
  __syncthreads();

  const int lane = tid & 31;
  const int w    = tid >> 5;
  const int lrow = lane & 15;
  const int half = lane >> 4;
  float* Abuf = sAbuf[w];
  float* Tbuf = sTbuf[w];

  const int nTiles = NNODES / 16;           // 3125, exact
  const int slots  = gridDim.x * WAVES;

  for (int tile = blockIdx.x * WAVES + w; tile < nTiles; tile += slots) {
    const int node = tile * 16 + lrow;
    const float4* hp = reinterpret_cast<const float4*>(h + (size_t)node * HDIM);
    const float4* mp = reinterpret_cast<const float4*>(hmsg + (size_t)node * HDIM);
#pragma unroll
    for (int i = 0; i < 8; ++i) {
      float4 a = hp[half * 8 + i];
      float4 b = mp[half * 8 + i];
      float* d = Abuf + lrow * TS + half * 32 + i * 4;
      d[0] = a.x + b.x; d[1] = a.y + b.y; d[2] = a.z + b.z; d[3] = a.w + b.w;
    }
    lds_fence();

    v8f acc[4];
    wave_gemm_16xK_x64<HDIM>(Abuf, TS, sW1p, sB1, acc, lrow, half);
    store_tile(Tbuf, TS, acc, lrow, half, /*silu=*/true);
    lds_fence();
    wave_gemm_16xK_x64<HDIM>(Tbuf, TS, sW2p, sB2, acc, lrow, half);
    lds_fence();
    store_tile(Tbuf, TS, acc, lrow, half, /*silu=*/false);
    lds_fence();

    if (!half) {
      float* rp = Tbuf + lrow * TS;
      float m = 0.f;
#pragma unroll
      for (int n = 0; n < HDIM; ++n) m += rp[n];
      m *= (1.0f / HDIM);
      float v = 0.f;
#pragma unroll
      for (int n = 0; n < HDIM; ++n) { float d = rp[n] - m; v += d * d; }
      v *= (1.0f / HDIM);
      const float rs = rsqrtf(v + 1e-5f);
      float* op = out + (size_t)node * HDIM;
#pragma unroll
      for (int n = 0; n < HDIM; ++n) op[n] = (rp[n] - m) * rs * sG[n] + sBe[n];
    }
    lds_fence();
  }
}

// ------------------------------------------------------------------
extern "C" void kernel_launch(void* const* d_in, const int* in_sizes, int n_in,
                              void* d_out, int out_size, void* d_ws, size_t ws_size,
                              hipStream_t stream) {
  const float* h    = (const float*)d_in[0];
  const float* pos  = (const float*)d_in[1];
  const int*   eidx = (const int*)d_in[2];
  const float* pe_w1 = (const float*)d_in[3];
  const float* pe_b1 = (const float*)d_in[4];
  const float* pe_w2 = (const float*)d_in[5];
  const float* pe_b2 = (const float*)d_in[6];
  const float* pe_g  = (const float*)d_in[7];
  const float* pe_be = (const float*)d_in[8];
  const float* ph_w1 = (const float*)d_in[9];
  const float* ph_b1 = (const float*)d_in[10];
  const float* ph_w2 = (const float*)d_in[11];
  const float* ph_b2 = (const float*)d_in[12];
  const float* ph_g  = (const float*)d_in[13];
  const float* ph_be = (const float*)d_in[14];
  const float* px_w1 = (const float*)d_in[15];
  const float* px_b1 = (const float*)d_in[16];
  const float* px_w2 = (const float*)d_in[17];
  const float* px_b2 = (const float*)d_in[18];

  float* out    = (float*)d_out;
  float* hupd   = out;                               // [N, 64]
  float* posOut = out + (size_t)NNODES * HDIM;       // [N, 3]
  float* hmsg   = (float*)d_ws;                      // [N, 64] accumulator

  const size_t nz1 = (size_t)NNODES * HDIM;
  const size_t nz2 = (size_t)NNODES * 3;
  egnn_zero_kernel<<<(int)((nz1 + 255) / 256), 256, 0, stream>>>(hmsg, nz1);
  egnn_zero_kernel<<<(int)((nz2 + 255) / 256), 256, 0, stream>>>(posOut, nz2);

  egnn_edge_kernel<<<1250, 128, 0, stream>>>(
      h, pos, eidx, pe_w1, pe_b1, pe_w2, pe_b2, pe_g, pe_be,
      px_w1, px_b1, px_w2, px_b2, hmsg, posOut);

  const int nodeTiles  = NNODES / 16;  // 3125
  const int nodeBlocks = (nodeTiles + WAVES - 1) / WAVES;  // 782
  egnn_node_kernel<<<nodeBlocks, 128, 0, stream>>>(
      h, hmsg, ph_w1, ph_b1, ph_w2, ph_b2, ph_g, ph_be, hupd);
}